// MoIELinear_26817775796689
// MI455X (gfx1250) — compile-verified
//
#include <hip/hip_runtime.h>
#include <cmath>

typedef __attribute__((ext_vector_type(2))) float v2f;
typedef __attribute__((ext_vector_type(8))) float v8f;
typedef int v4i __attribute__((vector_size(16)));            // GCC-vector flavor, matches builtin
typedef __attribute__((address_space(1))) v4i gv4i;          // global AS pointer pointee
typedef __attribute__((address_space(3))) v4i lv4i;          // LDS AS pointer pointee

#define IN_F  2048
#define OUT_F 2048
#define BLK_M 128
#define BLK_N 128
#define KT    16
#define KPAD  20          // 80B LDS row stride: 16B-aligned b128 stores, conflict-free b64 frag reads
#define EPS_  1e-8f

// ---- CDNA5 async global->LDS copy (ASYNCcnt-tracked), with portable fallback ----
static __device__ __forceinline__ void async_ld_b128(const void* g, void* l) {
#if defined(__has_builtin) && __has_builtin(__builtin_amdgcn_global_load_async_to_lds_b128)
    __builtin_amdgcn_global_load_async_to_lds_b128((gv4i*)g, (lv4i*)l, 0, 0);
#else
    *(float4*)l = *(const float4*)g;   // synchronous fallback
#endif
}
static __device__ __forceinline__ void wait_async0() {
#if defined(__has_builtin) && __has_builtin(__builtin_amdgcn_s_wait_asynccnt)
    __builtin_amdgcn_s_wait_asynccnt(0);
#else
    asm volatile("s_wait_asynccnt 0" ::: "memory");
#endif
}

__device__ __forceinline__ float softplusf(float v) {
    return (v > 20.0f) ? v : log1pf(expf(v));
}
__device__ __forceinline__ float sq4(float4 v) {
    return v.x * v.x + v.y * v.y + v.z * v.z + v.w * v.w;
}
__device__ __forceinline__ float block_reduce_sum(float v, float* red) {
    const int t = threadIdx.x;
    red[t] = v; __syncthreads();
#pragma unroll
    for (int off = 128; off > 0; off >>= 1) {
        if (t < off) red[t] += red[t + off];
        __syncthreads();
    }
    return red[0];
}

// keys = mu * softplus(sigma), fused per-row ||keys|| reduction. One block per out-row.
// Also warms L2 with mu/keys for the GEMM.
__global__ __launch_bounds__(256) void moie_keys_kernel(const float* __restrict__ mu,
                                                        const float* __restrict__ sg,
                                                        float* __restrict__ keys,
                                                        float* __restrict__ knorm) {
    __shared__ float red[256];
    const int row = blockIdx.x;
    const size_t base = (size_t)row * IN_F;
    float s = 0.f;
#pragma unroll
    for (int it = 0; it < IN_F / 4 / 256; ++it) {
        const int i = threadIdx.x + it * 256;
        float4 m = ((const float4*)(mu + base))[i];
        float4 g = ((const float4*)(sg + base))[i];
        float4 k;
        k.x = m.x * softplusf(g.x);
        k.y = m.y * softplusf(g.y);
        k.z = m.z * softplusf(g.z);
        k.w = m.w * softplusf(g.w);
        ((float4*)(keys + base))[i] = k;
        s += sq4(k);
    }
    s = block_reduce_sum(s, red);
    if (threadIdx.x == 0) knorm[row] = sqrtf(s);
}

// per-row ||x||; one block per row. Also warms L2 with x for the GEMM.
__global__ __launch_bounds__(256) void moie_xnorm_kernel(const float* __restrict__ x,
                                                         float* __restrict__ xnorm) {
    __shared__ float red[256];
    const int row = blockIdx.x;
    const float* p = x + (size_t)row * IN_F;
    float s = 0.f;
#pragma unroll
    for (int it = 0; it < IN_F / 4 / 256; ++it) {
        float4 v = ((const float4*)p)[threadIdx.x + it * 256];
        s += sq4(v);
    }
    s = block_reduce_sum(s, red);
    if (threadIdx.x == 0) xnorm[row] = sqrtf(s);
}

// Fused dual-GEMM (dots = x.keys^T, comp = x.mu^T) with double-buffered async
// LDS staging and f32 WMMA. 256 threads (8 wave32s), block tile 128x128, K-tile 16.
__global__ __launch_bounds__(256, 1) void moie_gemm_kernel(
        const float* __restrict__ x, const float* __restrict__ muW,
        const float* __restrict__ keys, const float* __restrict__ gate,
        const float* __restrict__ bias, const float* __restrict__ xnorm,
        const float* __restrict__ knorm, float* __restrict__ out,
        size_t scores_off, size_t masked_off) {

    __shared__ float As[2][BLK_M][KPAD];
    __shared__ float Bm[2][BLK_N][KPAD];
    __shared__ float Bk[2][BLK_N][KPAD];

    const int tid  = threadIdx.x;
    const int lane = tid & 31;
    const int wv   = tid >> 5;       // wave 0..7
    const int wM   = wv >> 2;        // 0..1 : 64-row slab
    const int wN   = wv & 3;         // 0..3 : 32-col slab
    const int l15  = lane & 15;
    const int lHi  = lane >> 4;      // 0/1 : K-pair select (A/B), M+8 select (C/D)

    const int rowBase = blockIdx.y * BLK_M;
    const int colBase = blockIdx.x * BLK_N;

    // staging map: thread owns 16B chunk (frow, fcol) and (frow+64, fcol) of each tile
    const int frow = tid >> 2;          // 0..63
    const int fcol = (tid & 3) << 2;    // 0,4,8,12

    const float* gx0 = &x[(size_t)(rowBase + frow) * IN_F + fcol];
    const float* gx1 = &x[(size_t)(rowBase + frow + 64) * IN_F + fcol];
    const float* gm0 = &muW[(size_t)(colBase + frow) * IN_F + fcol];
    const float* gm1 = &muW[(size_t)(colBase + frow + 64) * IN_F + fcol];
    const float* gq0 = &keys[(size_t)(colBase + frow) * IN_F + fcol];
    const float* gq1 = &keys[(size_t)(colBase + frow + 64) * IN_F + fcol];

    v8f accD[4][2];   // dots (x . keys^T)
    v8f accC[4][2];   // comp (x . mu^T)
    const v8f vzero = {0.f, 0.f, 0.f, 0.f, 0.f, 0.f, 0.f, 0.f};
#pragma unroll
    for (int mi = 0; mi < 4; ++mi)
#pragma unroll
        for (int ni = 0; ni < 2; ++ni) { accD[mi][ni] = vzero; accC[mi][ni] = vzero; }

    const int NST = IN_F / KT;   // 128 stages

    // prologue: stage 0 -> buffer 0
    {
        async_ld_b128(gx0, &As[0][frow][fcol]);
        async_ld_b128(gx1, &As[0][frow + 64][fcol]);
        async_ld_b128(gm0, &Bm[0][frow][fcol]);
        async_ld_b128(gm1, &Bm[0][frow + 64][fcol]);
        async_ld_b128(gq0, &Bk[0][frow][fcol]);
        async_ld_b128(gq1, &Bk[0][frow + 64][fcol]);
    }

    for (int s = 0; s < NST; ++s) {
        wait_async0();        // my stage-s copies landed in LDS
        __syncthreads();      // everyone's landed; everyone done reading buf[(s+1)&1]

        if (s + 1 < NST) {    // overlap stage-(s+1) DMA with stage-s WMMAs
            const int b = (s + 1) & 1;
            const int ko = (s + 1) * KT;
            async_ld_b128(gx0 + ko, &As[b][frow][fcol]);
            async_ld_b128(gx1 + ko, &As[b][frow + 64][fcol]);
            async_ld_b128(gm0 + ko, &Bm[b][frow][fcol]);
            async_ld_b128(gm1 + ko, &Bm[b][frow + 64][fcol]);
            async_ld_b128(gq0 + ko, &Bk[b][frow][fcol]);
            async_ld_b128(gq1 + ko, &Bk[b][frow + 64][fcol]);
        }

        const int b = s & 1;
#pragma unroll
        for (int kk = 0; kk < 4; ++kk) {
            const int c = kk * 4 + lHi * 2;
            v2f aF[4], bMF[2], bKF[2];
#pragma unroll
            for (int mi = 0; mi < 4; ++mi)
                aF[mi] = *(const v2f*)&As[b][wM * 64 + mi * 16 + l15][c];
#pragma unroll
            for (int ni = 0; ni < 2; ++ni) {
                bMF[ni] = *(const v2f*)&Bm[b][wN * 32 + ni * 16 + l15][c];
                bKF[ni] = *(const v2f*)&Bk[b][wN * 32 + ni * 16 + l15][c];
            }
#pragma unroll
            for (int mi = 0; mi < 4; ++mi)
#pragma unroll
                for (int ni = 0; ni < 2; ++ni) {
                    accD[mi][ni] = __builtin_amdgcn_wmma_f32_16x16x4_f32(
                        false, aF[mi], false, bKF[ni], (short)0, accD[mi][ni], false, false);
                    accC[mi][ni] = __builtin_amdgcn_wmma_f32_16x16x4_f32(
                        false, aF[mi], false, bMF[ni], (short)0, accC[mi][ni], false, false);
                }
        }
    }

    // ---- epilogue: scores / relu-gate / masked, 3 output planes ----
#pragma unroll
    for (int ni = 0; ni < 2; ++ni) {
        const int ng = colBase + wN * 32 + ni * 16 + l15;
        const float g  = gate[ng];
        const float bb = bias[ng];
        const float kn = knorm[ng];
#pragma unroll
        for (int mi = 0; mi < 4; ++mi) {
#pragma unroll
            for (int r = 0; r < 8; ++r) {
                const int mg = rowBase + wM * 64 + mi * 16 + lHi * 8 + r;
                const float xn = xnorm[mg];
                const float score  = accD[mi][ni][r] / fmaxf(xn * kn, EPS_);
                const float wgt    = fmaxf(score - g, 0.f);
                const float masked = (accC[mi][ni][r] + bb) * wgt;
                const size_t idx = (size_t)mg * OUT_F + ng;
                out[idx]              = masked;   // output plane (== masked)
                out[scores_off + idx] = score;    // scores plane
                out[masked_off + idx] = masked;   // masked plane
            }
        }
    }
}

extern "C" void kernel_launch(void* const* d_in, const int* in_sizes, int n_in,
                              void* d_out, int out_size, void* d_ws, size_t ws_size,
                              hipStream_t stream) {
    (void)n_in; (void)out_size; (void)ws_size;
    const float* x     = (const float*)d_in[0];
    const float* muW   = (const float*)d_in[1];
    const float* sigma = (const float*)d_in[2];
    const float* gate  = (const float*)d_in[3];
    const float* bias  = (const float*)d_in[4];
    float* out = (float*)d_out;

    float* keys  = (float*)d_ws;                  // 16 MiB
    float* knorm = keys + (size_t)OUT_F * IN_F;   // 2048 floats
    float* xnorm = knorm + OUT_F;                 // nrows floats

    const int nrows = in_sizes[0] / IN_F;         // 8192
    const size_t plane = (size_t)nrows * OUT_F;

    // precompute keys + norms (also warms the 192MB L2 for the GEMM)
    moie_keys_kernel<<<OUT_F, 256, 0, stream>>>(muW, sigma, keys, knorm);
    moie_xnorm_kernel<<<nrows, 256, 0, stream>>>(x, xnorm);

    dim3 grid(OUT_F / BLK_N, nrows / BLK_M);      // 16 x 64 = 1024 blocks
    moie_gemm_kernel<<<grid, 256, 0, stream>>>(x, muW, keys, gate, bias,
                                               xnorm, knorm, out, plane, 2 * plane);
}